// model_52922587021826
// MI455X (gfx1250) — compile-verified
//
#include <hip/hip_runtime.h>
#include <math.h>

// Problem constants (from reference)
#define T_DIM   4
#define B_DIM   16384
#define CS_DIM  8
#define NS_DIM  20
#define L_DIM   50000
#define K_DIM   128
#define NSAMP   (T_DIM * B_DIM)            // 65536 samples
#define WPB     8                          // waves per block (256 threads)
#define GROUPS  4                          // samples per wave (8 lanes each)
#define SPB     (WPB * GROUPS)             // samples per block = 32
#define LL_BLOCKS    (NSAMP / SPB)         // 2048
#define PRIOR_BLOCKS 4096
#define NTW ((size_t)L_DIM * K_DIM)        // 6,400,000 elems per timestep
#define NCW ((size_t)(L_DIM + 1) * K_DIM)  // 6,400,128 elems

typedef float v2f __attribute__((ext_vector_type(2)));
typedef float v8f __attribute__((ext_vector_type(8)));

// ---------------------------------------------------------------------------
// Wave32 full reduction using V_WMMA_F32_16X16X4_F32 with B == all-ones.
// A = (x, 0) per lane, B = 1 everywhere:  D[m][n] = x_m + x_{m+16} for every
// n, independent of B's lane layout and of which k-slot each lane owns.
// Summing a lane's 8 D VGPRs covers 8 distinct rows of one half; one xor-16
// add then yields the full 32-lane sum in every lane.
// ---------------------------------------------------------------------------
__device__ __forceinline__ float wave_reduce_wmma(float x) {
    v2f a; a[0] = x;   a[1] = 0.0f;
    v2f b; b[0] = 1.0f; b[1] = 1.0f;
    v8f c = {0.f, 0.f, 0.f, 0.f, 0.f, 0.f, 0.f, 0.f};
    v8f d = __builtin_amdgcn_wmma_f32_16x16x4_f32(
        /*neg_a=*/false, a, /*neg_b=*/false, b,
        /*c_mod=*/(short)0, c, /*reuse_a=*/false, /*reuse_b=*/false);
    float s = d[0] + d[1] + d[2] + d[3] + d[4] + d[5] + d[6] + d[7];
    s += __shfl_xor(s, 16, 32);
    return s;
}

__device__ __forceinline__ float softplusf(float x) {
    // numerically stable log(1 + exp(x))
    return fmaxf(x, 0.0f) + log1pf(expf(-fabsf(x)));
}

__device__ __forceinline__ float dot4(const float4 a, const float4 b) {
    return a.x * b.x + a.y * b.y + a.z * b.z + a.w * b.w;
}

// ---------------------------------------------------------------------------
// Kernel 1: likelihood.  4 samples per wave; each sample owned by an 8-lane
// group, lane j covering float4-chunks {j, j+8, j+16, j+24} of K=128 so each
// of the 4 row loads is a contiguous 128B request per group (512B/wave instr,
// L2-resident).  Per-eta reduction = 3 shfl_xor rounds confined to the group.
// ---------------------------------------------------------------------------
__global__ __launch_bounds__(256) void ll_kernel(
    const int*   __restrict__ contexts,   // [T,B,CS]
    const int*   __restrict__ targets,    // [T,B,1]
    const int*   __restrict__ neg_idx,    // [T,B,NS]
    const float* __restrict__ cw,         // [L+1,K]
    const float* __restrict__ tw,         // [T,L,K]
    float*       __restrict__ out_partial)// [LL_BLOCKS]
{
    __shared__ float sacc[SPB];
    const int lane = threadIdx.x & 31;
    const int wid  = threadIdx.x >> 5;
    const int g    = lane >> 3;                    // group (sample) in wave
    const int j    = lane & 7;                     // lane within group
    const int s    = (blockIdx.x * WPB + wid) * GROUPS + g;  // sample id
    const int t    = s >> 14;                      // B_DIM == 2^14

    // context window sum: 8 gathered rows of context_weight
    float4 c0 = make_float4(0.f,0.f,0.f,0.f), c1 = c0, c2 = c0, c3 = c0;
    const int* ci = contexts + (size_t)s * CS_DIM;
#pragma unroll
    for (int i = 0; i < CS_DIM; ++i) {
        const float4* row = reinterpret_cast<const float4*>(
            cw + (size_t)ci[i] * K_DIM);           // idx may be L (zero row)
        const float4 v0 = row[j];
        const float4 v1 = row[j + 8];
        const float4 v2 = row[j + 16];
        const float4 v3 = row[j + 24];
        c0.x += v0.x; c0.y += v0.y; c0.z += v0.z; c0.w += v0.w;
        c1.x += v1.x; c1.y += v1.y; c1.z += v1.z; c1.w += v1.w;
        c2.x += v2.x; c2.y += v2.y; c2.z += v2.z; c2.w += v2.w;
        c3.x += v3.x; c3.y += v3.y; c3.z += v3.z; c3.w += v3.w;
    }

    const float* twt = tw + (size_t)t * NTW;
    float acc;
    {   // positive target (label 1): softplus(-eta)
        const float4* row = reinterpret_cast<const float4*>(
            twt + (size_t)targets[s] * K_DIM);
        float p = dot4(row[j], c0) + dot4(row[j + 8], c1)
                + dot4(row[j + 16], c2) + dot4(row[j + 24], c3);
        p += __shfl_xor(p, 1, 32);
        p += __shfl_xor(p, 2, 32);
        p += __shfl_xor(p, 4, 32);                 // eta in all 8 group lanes
        acc = softplusf(-p);
    }
    const int* ni = neg_idx + (size_t)s * NS_DIM;
    for (int n = 0; n < NS_DIM; ++n) {             // negatives: softplus(+eta)
        const float4* row = reinterpret_cast<const float4*>(
            twt + (size_t)ni[n] * K_DIM);
        float p = dot4(row[j], c0) + dot4(row[j + 8], c1)
                + dot4(row[j + 16], c2) + dot4(row[j + 24], c3);
        p += __shfl_xor(p, 1, 32);
        p += __shfl_xor(p, 2, 32);
        p += __shfl_xor(p, 4, 32);
        acc += softplusf(p);
    }

    if (j == 0) sacc[wid * GROUPS + g] = acc;      // acc uniform in group
    __syncthreads();
    if (threadIdx.x == 0) {
        float bsum = 0.f;
#pragma unroll
        for (int i = 0; i < SPB; ++i) bsum += sacc[i];
        out_partial[blockIdx.x] = bsum;
    }
}

// ---------------------------------------------------------------------------
// Kernel 2: Gaussian prior quadratic terms (constants folded analytically in
// finalize).  sig=1 terms: -0.5 x^2 ; sig_loc=0.01 terms: -5000 d^2.
// ---------------------------------------------------------------------------
__global__ __launch_bounds__(256) void prior_kernel(
    const float* __restrict__ cw,          // [L+1,K]
    const float* __restrict__ tw,          // [T,L,K]
    float*       __restrict__ out_partial) // [PRIOR_BLOCKS]
{
    const size_t tid    = (size_t)blockIdx.x * blockDim.x + threadIdx.x;
    const size_t stride = (size_t)gridDim.x * blockDim.x;

    float part = 0.0f;
    for (size_t i = tid; i < NCW; i += stride) {
        const float x = cw[i];
        part -= 0.5f * x * x;
    }
    for (size_t i = tid; i < NTW; i += stride) {
        const float w0 = tw[i];
        const float w1 = tw[NTW + i];
        const float w2 = tw[2 * NTW + i];
        const float w3 = tw[3 * NTW + i];
        const float d0 = w0 - w3;                  // tw[0]   - tw[-1]
        const float d1 = w1 - w0;                  // tw[1]   - tw[0]
        const float d2 = w2 - w1;
        const float d3 = w3 - w2;
        part += -0.5f * w3 * w3
                - 5000.0f * (d0 * d0 + d1 * d1 + d2 * d2 + d3 * d3);
    }

    const float wsum = wave_reduce_wmma(part);     // EXEC all-ones here
    __shared__ float sw[WPB];
    const int lane = threadIdx.x & 31;
    const int wid  = threadIdx.x >> 5;
    if (lane == 0) sw[wid] = wsum;
    __syncthreads();
    if (threadIdx.x == 0) {
        float b = 0.f;
#pragma unroll
        for (int i = 0; i < WPB; ++i) b += sw[i];
        out_partial[blockIdx.x] = b;
    }
}

// ---------------------------------------------------------------------------
// Kernel 3: final reduction + closed-form prior constants + loss assembly.
// ---------------------------------------------------------------------------
__global__ __launch_bounds__(256) void finalize_kernel(
    const float* __restrict__ ll_part,   // [LL_BLOCKS]
    const float* __restrict__ pr_part,   // [PRIOR_BLOCKS]
    float*       __restrict__ out)       // [1]
{
    float pll = 0.f, ppr = 0.f;
    for (int i = threadIdx.x; i < LL_BLOCKS; i += 256)    pll += ll_part[i];
    for (int i = threadIdx.x; i < PRIOR_BLOCKS; i += 256) ppr += pr_part[i];

    const float wll = wave_reduce_wmma(pll);
    const float wpr = wave_reduce_wmma(ppr);

    __shared__ float sll[WPB], spr[WPB];
    const int lane = threadIdx.x & 31;
    const int wid  = threadIdx.x >> 5;
    if (lane == 0) { sll[wid] = wll; spr[wid] = wpr; }
    __syncthreads();

    if (threadIdx.x == 0) {
        double all = 0.0, apr = 0.0;
#pragma unroll
        for (int i = 0; i < WPB; ++i) { all += (double)sll[i]; apr += (double)spr[i]; }
        // Normal(0,sig).log_prob constants:
        //   sig=1    : per-elem const = -(0.5*ln(2*pi))            = -C1
        //   sig=0.01 : per-elem const = -(ln(0.01) + 0.5*ln(2*pi)) = -C2
        const double C1 = 0.91893853320467274178;      // 0.5*ln(2*pi)
        const double C2 = C1 - 4.6051701859880913680;  // + ln(0.01)
        const double n_sig1 = (double)NCW + (double)NTW;   // cw + tw[-1]
        const double n_loc  = 4.0 * (double)NTW;           // 4 diff groups
        const double log_prior = apr - n_sig1 * C1 - n_loc * C2;
        // ll_pos+ll_neg == -all  =>  loss = N_EPOCHS*all - log_prior
        const double loss = 1000.0 * all - log_prior;
        out[0] = (float)loss;
    }
}

// ---------------------------------------------------------------------------
extern "C" void kernel_launch(void* const* d_in, const int* in_sizes, int n_in,
                              void* d_out, int out_size, void* d_ws, size_t ws_size,
                              hipStream_t stream)
{
    (void)in_sizes; (void)n_in; (void)out_size; (void)ws_size;
    const int*   contexts = (const int*)d_in[0];
    const int*   targets  = (const int*)d_in[1];
    const int*   neg_idx  = (const int*)d_in[2];
    const float* cw       = (const float*)d_in[3];
    const float* tw       = (const float*)d_in[4];
    float* out   = (float*)d_out;
    float* ws_ll = (float*)d_ws;             // 2048 floats
    float* ws_pr = ws_ll + LL_BLOCKS;        // 4096 floats (24 KB total)

    ll_kernel<<<LL_BLOCKS, 256, 0, stream>>>(contexts, targets, neg_idx, cw, tw, ws_ll);
    prior_kernel<<<PRIOR_BLOCKS, 256, 0, stream>>>(cw, tw, ws_pr);
    finalize_kernel<<<1, 256, 0, stream>>>(ws_ll, ws_pr, out);
}